// BiRNNLM_90598040142180
// MI455X (gfx1250) — compile-verified
//
#include <hip/hip_runtime.h>
#include <hip/hip_bf16.h>
#include <math.h>

#define V_SZ 32000
#define E_SZ 32
#define H_SZ 8
#define S_SZ 128
#define B_SZ 64
#define NROW (S_SZ * B_SZ)      // 8192 GEMM rows
#define NTILE (V_SZ / 16)       // 2000 column tiles
#define WAVES 8

typedef __attribute__((ext_vector_type(2))) float v2f;
typedef __attribute__((ext_vector_type(8))) float v8f;

// ---------------------------------------------------------------------------
// Kernel 1: the two tiny recurrent scans. blockIdx.x = 0 (fwd) / 1 (bwd),
// one thread per batch column, hidden state in registers.
// Stores the PRE-update hidden state at each step (matches lax.scan's ys).
// ---------------------------------------------------------------------------
__global__ __launch_bounds__(B_SZ) void rnn_scan_kernel(
    const int* __restrict__ idx, const float* __restrict__ lookup,
    const float* __restrict__ Wx_f, const float* __restrict__ Wh_f,
    const float* __restrict__ Wx_b, const float* __restrict__ Wh_b,
    const float* __restrict__ H0_f, const float* __restrict__ H0_b,
    const float* __restrict__ b_f1, const float* __restrict__ b_f2,
    const float* __restrict__ b_b1, const float* __restrict__ b_b2,
    float* __restrict__ ws_f, float* __restrict__ ws_b)
{
    const int dir = blockIdx.x;     // 0 = forward, 1 = backward
    __shared__ float Wx[E_SZ * H_SZ];
    __shared__ float Wh[H_SZ * H_SZ];
    __shared__ float bias[H_SZ];

    const float* Wxg = dir ? Wx_b : Wx_f;
    const float* Whg = dir ? Wh_b : Wh_f;
    const float* bg1 = dir ? b_b1 : b_f1;
    const float* bg2 = dir ? b_b2 : b_f2;
    const float* H0  = dir ? H0_b : H0_f;
    float*       tab = dir ? ws_b : ws_f;

    const int t = threadIdx.x;
    for (int i = t; i < E_SZ * H_SZ; i += B_SZ) Wx[i] = Wxg[i];
    for (int i = t; i < H_SZ * H_SZ; i += B_SZ) Wh[i] = Whg[i];
    if (t < H_SZ) bias[t] = bg1[t] + bg2[t];
    __syncthreads();

    const int b = t;                // batch column
    float h[H_SZ];
#pragma unroll
    for (int j = 0; j < H_SZ; ++j) h[j] = H0[j];

    for (int step = 0; step < S_SZ; ++step) {
        const int s = dir ? (S_SZ - 1 - step) : step;
        // store pre-update state: Hf_table[s] / Hb_table[s]
#pragma unroll
        for (int j = 0; j < H_SZ; ++j) tab[(s * B_SZ + b) * H_SZ + j] = h[j];

        const int token = idx[s * B_SZ + b];
        const float* x = lookup + (long long)token * E_SZ;
        float xr[E_SZ];
#pragma unroll
        for (int e = 0; e < E_SZ; ++e) xr[e] = x[e];

        float hn[H_SZ];
#pragma unroll
        for (int j = 0; j < H_SZ; ++j) {
            float acc = bias[j];
#pragma unroll
            for (int e = 0; e < E_SZ; ++e) acc += xr[e] * Wx[e * H_SZ + j];
#pragma unroll
            for (int k = 0; k < H_SZ; ++k) acc += h[k] * Wh[k * H_SZ + j];
            hn[j] = tanhf(acc);
        }
#pragma unroll
        for (int j = 0; j < H_SZ; ++j) h[j] = hn[j];
    }
}

// ---------------------------------------------------------------------------
// Kernel 2: logits GEMM (K=16) via chained V_WMMA_F32_16X16X4_F32 + fused
// log_softmax. One block = 16 GEMM rows; 8 waves stripe the 2000 vocab tiles.
// Pass 1: online max/sum-exp (weight_o stays in L2). Pass 2: recompute and
// write  logit - (max + log(sum)).  HBM traffic ~= the 1.05 GB output write.
// ---------------------------------------------------------------------------
__global__ __launch_bounds__(32 * WAVES) void logits_lsm_kernel(
    const float* __restrict__ ws_f, const float* __restrict__ ws_b,
    const float* __restrict__ Wo, const float* __restrict__ bo,
    float* __restrict__ out)
{
    __shared__ float Acat[16][16];
    __shared__ float m_sh[WAVES][16];
    __shared__ float s_sh[WAVES][16];
    __shared__ float Lrow[16];

    const int tid   = threadIdx.x;
    const int Rbase = blockIdx.x * 16;

    // Stage the 16x16 Hcat tile: k<8 from forward table, k>=8 from backward.
    {
        const int r = tid >> 4, k = tid & 15;
        const int R = Rbase + r;
        Acat[r][k] = (k < 8) ? ws_f[R * H_SZ + k] : ws_b[R * H_SZ + (k - 8)];
    }
    __syncthreads();

    const int lane = tid & 31;
    const int wid  = tid >> 5;
    const int hi   = lane >> 4;     // 0: rows 0-7 of tile, 1: rows 8-15
    const int lo   = lane & 15;     // tile column

    // A-matrix per ISA layout: VGPR0=K even, VGPR1=K odd; lanes 16-31 get K+2.
    v2f a[4];
#pragma unroll
    for (int g = 0; g < 4; ++g) {
        a[g].x = Acat[lo][4 * g + 2 * hi];
        a[g].y = Acat[lo][4 * g + 2 * hi + 1];
    }

    // B-matrix base: lane reads weight_o rows (4g+2*hi, 4g+1+2*hi), col n0+lo.
    const float* wbase = Wo + (long long)(2 * hi) * V_SZ + lo;

    float m[8], sum[8];
#pragma unroll
    for (int i = 0; i < 8; ++i) { m[i] = -INFINITY; sum[i] = 0.f; }

    // ---------------- pass 1: online logsumexp ----------------
    for (int t = wid; t < NTILE; t += WAVES) {
        const int n0 = t * 16;
        v8f c = {};
#pragma unroll
        for (int g = 0; g < 4; ++g) {
            v2f bm;
            bm.x = wbase[n0 + (4 * g) * V_SZ];
            bm.y = wbase[n0 + (4 * g + 1) * V_SZ];
            c = __builtin_amdgcn_wmma_f32_16x16x4_f32(
                    false, a[g], false, bm, (short)0, c, false, false);
        }
        const float bocol = bo[n0 + lo];
#pragma unroll
        for (int i = 0; i < 8; ++i) {
            const float v  = c[i] + bocol;
            const float mn = fmaxf(m[i], v);
            sum[i] = sum[i] * __expf(m[i] - mn) + __expf(v - mn);
            m[i]   = mn;
        }
    }

    // reduce across the 16 lanes sharing the same row group
#pragma unroll
    for (int mask = 1; mask <= 8; mask <<= 1) {
#pragma unroll
        for (int i = 0; i < 8; ++i) {
            const float m2 = __shfl_xor(m[i], mask, 32);
            const float s2 = __shfl_xor(sum[i], mask, 32);
            const float mn = fmaxf(m[i], m2);
            sum[i] = sum[i] * __expf(m[i] - mn) + s2 * __expf(m2 - mn);
            m[i]   = mn;
        }
    }
    if (lo == 0) {
#pragma unroll
        for (int i = 0; i < 8; ++i) {
            m_sh[wid][8 * hi + i] = m[i];
            s_sh[wid][8 * hi + i] = sum[i];
        }
    }
    __syncthreads();
    if (tid < 16) {
        float mm = -INFINITY, ss = 0.f;
        for (int w = 0; w < WAVES; ++w) {
            const float m2 = m_sh[w][tid];
            const float s2 = s_sh[w][tid];
            const float mn = fmaxf(mm, m2);
            ss = ss * __expf(mm - mn) + s2 * __expf(m2 - mn);
            mm = mn;
        }
        Lrow[tid] = mm + __logf(ss);
    }
    __syncthreads();

    float L[8];
#pragma unroll
    for (int i = 0; i < 8; ++i) L[i] = Lrow[8 * hi + i];

    // ---------------- pass 2: recompute + write ----------------
    for (int t = wid; t < NTILE; t += WAVES) {
        const int n0 = t * 16;
        v8f c = {};
#pragma unroll
        for (int g = 0; g < 4; ++g) {
            v2f bm;
            bm.x = wbase[n0 + (4 * g) * V_SZ];
            bm.y = wbase[n0 + (4 * g + 1) * V_SZ];
            c = __builtin_amdgcn_wmma_f32_16x16x4_f32(
                    false, a[g], false, bm, (short)0, c, false, false);
        }
        const float bocol = bo[n0 + lo];
#pragma unroll
        for (int i = 0; i < 8; ++i) {
            const long long row = Rbase + 8 * hi + i;
            out[row * (long long)V_SZ + n0 + lo] = c[i] + bocol - L[i];
        }
    }
}

// ---------------------------------------------------------------------------
extern "C" void kernel_launch(void* const* d_in, const int* in_sizes, int n_in,
                              void* d_out, int out_size, void* d_ws, size_t ws_size,
                              hipStream_t stream) {
    (void)in_sizes; (void)n_in; (void)out_size; (void)ws_size;
    const int*   input_batch = (const int*)  d_in[0];
    const float* lookup      = (const float*)d_in[1];
    const float* weight_xf   = (const float*)d_in[2];
    const float* weight_hf   = (const float*)d_in[3];
    const float* weight_xb   = (const float*)d_in[4];
    const float* weight_hb   = (const float*)d_in[5];
    const float* weight_o    = (const float*)d_in[6];
    const float* H_f         = (const float*)d_in[7];
    const float* H_b         = (const float*)d_in[8];
    const float* b_f1        = (const float*)d_in[9];
    const float* b_f2        = (const float*)d_in[10];
    const float* b_b1        = (const float*)d_in[11];
    const float* b_b2        = (const float*)d_in[12];
    const float* b_o         = (const float*)d_in[13];

    float* ws_f = (float*)d_ws;                  // S*B*H floats
    float* ws_b = ws_f + (size_t)NROW * H_SZ;    // S*B*H floats

    rnn_scan_kernel<<<2, B_SZ, 0, stream>>>(
        input_batch, lookup, weight_xf, weight_hf, weight_xb, weight_hb,
        H_f, H_b, b_f1, b_f2, b_b1, b_b2, ws_f, ws_b);

    logits_lsm_kernel<<<NROW / 16, 32 * WAVES, 0, stream>>>(
        ws_f, ws_b, weight_o, b_o, (float*)d_out);
}